// Attention_Module_36197984370957
// MI455X (gfx1250) — compile-verified
//
#include <hip/hip_runtime.h>
#include <math.h>
#include <stdint.h>

// ---- problem constants (match reference) ----
#define BSZ   32
#define SEQ   2048
#define HID   1024
#define DOUT  1024            // encoder feature dim == OUT
#define CATK  (HID + DOUT)    // 2048
#define SPLIT 8               // S-splits per batch (32*8 = 256 blocks for BW)
#define ROWS_PER_SPLIT (SEQ / SPLIT)  // 256
#define CHUNK 8               // rows staged in LDS per iteration

typedef __attribute__((ext_vector_type(2))) float v2f;
typedef __attribute__((ext_vector_type(4))) float v4f;
typedef __attribute__((ext_vector_type(8))) float v8f;

// CDNA5 async global->LDS copy (16B per lane), tracked by ASYNCcnt.
// VDST operand = per-lane LDS byte address, VADDR = per-lane 64-bit global addr.
__device__ __forceinline__ void async_copy_b128(uint32_t lds_addr, const void* gaddr) {
    asm volatile("global_load_async_to_lds_b128 %0, %1, off"
                 :: "v"(lds_addr), "v"(gaddr)
                 : "memory");
}
__device__ __forceinline__ void wait_async_le8() {
    asm volatile("s_wait_asynccnt 0x8" ::: "memory");
}
__device__ __forceinline__ void wait_async_0() {
    asm volatile("s_wait_asynccnt 0x0" ::: "memory");
}

// ---------------------------------------------------------------------------
// C[32,N] = act( A[32,K] @ B[N,K]^T )  via V_WMMA_F32_16X16X4_F32 (fp32 WMMA).
// One wave per 16x16 output tile; 8 waves / block.
// A-frag per lane l:  row = m0 + (l&15), holds K = k+2*(l>>4), k+2*(l>>4)+1
// B-frag per lane l:  col = n0 + (l&15), same K pair (B row-major [N,K] => B^T)
// D per lane l, vgpr i: M = m0 + i + 8*(l>>4), N = n0 + (l&15)
// ---------------------------------------------------------------------------
template <int KDIM, bool TANH>
__global__ __launch_bounds__(256) void wmma_gemm32(const float* __restrict__ A,
                                                   const float* __restrict__ B,
                                                   float* __restrict__ C,
                                                   int N) {
    const int lane = threadIdx.x & 31;
    const int wave = threadIdx.x >> 5;
    const int half = lane >> 4;     // 0: K=k..k+1, 1: K=k+2..k+3
    const int lo   = lane & 15;
    const int n0   = (blockIdx.x * 8 + wave) * 16;
    const int m0   = blockIdx.y * 16;

    const float* ap = A + (size_t)(m0 + lo) * KDIM + 2 * half;
    const float* bp = B + (size_t)(n0 + lo) * KDIM + 2 * half;

    v8f c = {};
#pragma unroll 8
    for (int k = 0; k < KDIM; k += 4) {
        v2f a = *(const v2f*)(ap + k);
        v2f b = *(const v2f*)(bp + k);
        // (neg_a, A, neg_b, B, c_mod, C, reuse_a, reuse_b)
        c = __builtin_amdgcn_wmma_f32_16x16x4_f32(false, a, false, b,
                                                  (short)0, c, false, false);
    }
#pragma unroll
    for (int i = 0; i < 8; ++i) {
        int row = m0 + i + 8 * half;
        float v = c[i];
        if (TANH) v = tanhf(v);
        C[(size_t)row * N + n0 + lo] = v;
    }
}

// ---------------------------------------------------------------------------
// Kernel 2a: single pass over encoder_outs for one (batch, S-split):
// scores (masked) + online-softmax partial (m, l, acc[1024]).
// Global->LDS staging via async copies with double buffering.
// Dynamic LDS layout (floats):
//   q_s    [1024]
//   rows0  [CHUNK*1024]
//   rows1  [CHUNK*1024]
//   sc     [CHUNK]
// ---------------------------------------------------------------------------
__global__ __launch_bounds__(256) void attn_scores_partial(
    const float* __restrict__ enc,      // [B, SEQ, DOUT]
    const float* __restrict__ q,        // [B, DOUT]
    const int*   __restrict__ src_lens, // [B]
    float* __restrict__ scores_g,       // [B, SEQ]
    float* __restrict__ m_g,            // [B*SPLIT]
    float* __restrict__ l_g,            // [B*SPLIT]
    float* __restrict__ acc_g)          // [B*SPLIT, DOUT]
{
    extern __shared__ float smem[];
    float* q_s  = smem;                          // 1024
    float* rows[2];
    rows[0]     = smem + DOUT;                   // 8192
    rows[1]     = rows[0] + CHUNK * DOUT;        // 8192
    float* sc   = rows[1] + CHUNK * DOUT;        // 8

    const int b     = blockIdx.x;
    const int split = blockIdx.y;
    const int tid   = threadIdx.x;
    const int lane  = tid & 31;
    const int wave  = tid >> 5;
    const int len   = src_lens[b];
    const int sbase = split * ROWS_PER_SPLIT;
    const float* ebase = enc + ((size_t)b * SEQ + sbase) * DOUT;

    const uint32_t lds_rows0 = (uint32_t)(uint64_t)rows[0];
    const uint32_t lds_rows1 = (uint32_t)(uint64_t)rows[1];

    for (int i = tid; i < DOUT; i += 256) q_s[i] = q[b * DOUT + i];

    // preload chunk 0 into buffer 0 (8 x b128 per lane => 8 ASYNCcnt per wave)
    {
        const v4f* src = (const v4f*)ebase;
#pragma unroll
        for (int j = 0; j < 8; ++j) {
            int idx = tid + j * 256;             // 2048 x 16B = 32KB
            async_copy_b128(lds_rows0 + idx * 16, src + idx);
        }
    }

    float m = -INFINITY, l = 0.0f;
    float acc0 = 0.f, acc1 = 0.f, acc2 = 0.f, acc3 = 0.f;

    const int nchunks = ROWS_PER_SPLIT / CHUNK;  // 32
    for (int ci = 0; ci < nchunks; ++ci) {
        const int cur = ci & 1;

        // prefetch next chunk into the other buffer, then wait for current
        if (ci + 1 < nchunks) {
            const v4f* src = (const v4f*)(ebase + (size_t)(ci + 1) * CHUNK * DOUT);
            const uint32_t dst = cur ? lds_rows0 : lds_rows1;
#pragma unroll
            for (int j = 0; j < 8; ++j) {
                int idx = tid + j * 256;
                async_copy_b128(dst + idx * 16, src + idx);
            }
            wait_async_le8();                    // current buffer complete
        } else {
            wait_async_0();
        }
        __syncthreads();                         // whole block sees buffer

        const float* rbuf = rows[cur];

        // wave w: dot(rbuf[w], q) with wave32 xor-reduction
        {
            const float* r = rbuf + wave * DOUT;
            float p = 0.f;
#pragma unroll 8
            for (int d = lane; d < DOUT; d += 32) p += r[d] * q_s[d];
            p += __shfl_xor(p, 16);
            p += __shfl_xor(p, 8);
            p += __shfl_xor(p, 4);
            p += __shfl_xor(p, 2);
            p += __shfl_xor(p, 1);
            if (lane == 0) {
                int s = sbase + ci * CHUNK + wave;
                float masked = (s < len) ? p : 0.0f;
                if (masked == 0.0f) masked = -1e10f;   // torch mask semantics
                sc[wave] = masked;
                scores_g[b * SEQ + s] = masked;
            }
        }
        __syncthreads();

        // online softmax update (redundant scalar math, per-thread)
        float newm = m;
#pragma unroll
        for (int i = 0; i < CHUNK; ++i) newm = fmaxf(newm, sc[i]);
        float r = __expf(m - newm);     // m==-inf -> r==0, l==0, acc==0: safe
        float pv[CHUNK];
        float psum = 0.f;
#pragma unroll
        for (int i = 0; i < CHUNK; ++i) { pv[i] = __expf(sc[i] - newm); psum += pv[i]; }
        l = l * r + psum;
        m = newm;
        acc0 *= r; acc1 *= r; acc2 *= r; acc3 *= r;
#pragma unroll
        for (int i = 0; i < CHUNK; ++i) {
            const float* rr = rbuf + i * DOUT;
            float p = pv[i];
            acc0 += p * rr[tid];
            acc1 += p * rr[tid + 256];
            acc2 += p * rr[tid + 512];
            acc3 += p * rr[tid + 768];
        }
        __syncthreads();                // done reading buf before it is refilled
    }

    const int pidx = b * SPLIT + split;
    if (tid == 0) { m_g[pidx] = m; l_g[pidx] = l; }
    float* ag = acc_g + (size_t)pidx * DOUT;
    ag[tid]       = acc0;
    ag[tid + 256] = acc1;
    ag[tid + 512] = acc2;
    ag[tid + 768] = acc3;
}

// ---------------------------------------------------------------------------
// Kernel 2b: flash-merge SPLIT partials per batch; write cat=[ctx|hidden]
// and the transposed attn[S,B] output.
// ---------------------------------------------------------------------------
__global__ __launch_bounds__(256) void attn_combine(
    const float* __restrict__ scores_g,
    const float* __restrict__ m_g,
    const float* __restrict__ l_g,
    const float* __restrict__ acc_g,
    const float* __restrict__ hidden,   // [B, HID]
    float* __restrict__ cat,            // [B, CATK]
    float* __restrict__ attn_out)       // [SEQ, B]
{
    const int b   = blockIdx.x;
    const int tid = threadIdx.x;

    float M = -INFINITY;
#pragma unroll
    for (int i = 0; i < SPLIT; ++i) M = fmaxf(M, m_g[b * SPLIT + i]);
    float w[SPLIT];
    float L = 0.f;
#pragma unroll
    for (int i = 0; i < SPLIT; ++i) {
        w[i] = __expf(m_g[b * SPLIT + i] - M);   // masked-only partials -> 0
        L += w[i] * l_g[b * SPLIT + i];
    }
    const float invL = 1.0f / L;

#pragma unroll
    for (int j = 0; j < 4; ++j) {
        int d = tid + j * 256;
        float s = 0.f;
#pragma unroll
        for (int i = 0; i < SPLIT; ++i)
            s += w[i] * acc_g[(size_t)(b * SPLIT + i) * DOUT + d];
        cat[(size_t)b * CATK + d]       = s * invL;                 // ctx
        cat[(size_t)b * CATK + HID + d] = hidden[b * HID + d];      // hidden
    }

    for (int s = tid; s < SEQ; s += 256)
        attn_out[(size_t)s * BSZ + b] =
            __expf(scores_g[b * SEQ + s] - M) * invL;
}

// ---------------------------------------------------------------------------
extern "C" void kernel_launch(void* const* d_in, const int* in_sizes, int n_in,
                              void* d_out, int out_size, void* d_ws, size_t ws_size,
                              hipStream_t stream) {
    const float* hidden   = (const float*)d_in[0];   // [32,1024]
    const float* enc      = (const float*)d_in[1];   // [32,2048,1024]
    const int*   src_lens = (const int*)  d_in[2];   // [32]
    const float* W1       = (const float*)d_in[3];   // [1024,1024]
    const float* W2       = (const float*)d_in[4];   // [1024,2048]

    float* out      = (float*)d_out;                 // [32,1024]
    float* attn_out = out + BSZ * DOUT;              // [2048,32]

    // workspace layout (floats)
    float* ws       = (float*)d_ws;
    float* q        = ws;                  // 32768
    float* cat      = ws + 32768;          // 65536
    float* scores_g = ws + 98304;          // 65536
    float* m_g      = ws + 163840;         // 256
    float* l_g      = ws + 164096;         // 256
    float* acc_g    = ws + 164352;         // 262144  (total ~1.63 MB)

    // 1) q = hidden @ W1^T   (fp32 WMMA)
    wmma_gemm32<HID, false><<<dim3(8, 2), 256, 0, stream>>>(hidden, W1, q, DOUT);

    // 2) single pass over encoder_outs: scores + flash partials
    //    dynamic LDS: q(4KB) + 2 x 32KB row buffers + CHUNK floats
    const size_t lds_bytes = (DOUT + 2 * CHUNK * DOUT + CHUNK) * sizeof(float);
    attn_scores_partial<<<dim3(BSZ, SPLIT), 256, lds_bytes, stream>>>(
        enc, q, src_lens, scores_g, m_g, l_g, acc_g);

    // 3) merge partials -> ctx, cat, attn^T
    attn_combine<<<BSZ, 256, 0, stream>>>(scores_g, m_g, l_g, acc_g,
                                          hidden, cat, attn_out);

    // 4) out = tanh(cat @ W2^T)  (fp32 WMMA)
    wmma_gemm32<CATK, true><<<dim3(8, 2), 256, 0, stream>>>(cat, W2, out, DOUT);
}